// SIN_LDS_37271726195456
// MI455X (gfx1250) — compile-verified
//
#include <hip/hip_runtime.h>
#include <hip/hip_bf16.h>
#include <math.h>

// LDS (linear dynamical system) forward filter / backward smoother / sampler.
// D=16 matches the CDNA5 WMMA tile exactly: all 16x16x16 fp32 matmuls are done
// with 4 chained V_WMMA_F32_16X16X4_F32. One wave32 per batch chain (B=64).
// Backward pass double-buffers its saved filter-state tiles into LDS with
// GLOBAL_LOAD_ASYNC_TO_LDS_B128 (ASYNCcnt) to hide HBM latency on the serial chain.

#define DN 16
#define BN 64
#define TN 1000
#define LDG 33   // Gauss-Jordan augmented stride (16 | 16 | pad)
#define LDP 17   // padded LDS matrix stride
#define JIT 1e-6f

typedef __attribute__((ext_vector_type(2))) float v2f;
typedef __attribute__((ext_vector_type(8))) float v8f;

__device__ __forceinline__ int lane_id() { return (int)(threadIdx.x & 31u); }

__device__ __forceinline__ v8f vzero8() {
  v8f c;
#pragma unroll
  for (int r = 0; r < 8; ++r) c[r] = 0.0f;
  return c;
}

// per-lane 0/1 mask for diagonal elements in the WMMA C-fragment layout
__device__ __forceinline__ v8f diag_mask() {
  const int l = lane_id(), half = l >> 4, lm = l & 15;
  v8f d;
#pragma unroll
  for (int r = 0; r < 8; ++r) d[r] = (r + 8 * half == lm) ? 1.0f : 0.0f;
  return d;
}

// ---- WMMA fragment helpers (16x16 f32 C/D layout: lane l holds (M=r+8*(l>>4), N=l&15)) ----
__device__ __forceinline__ void frag_store(float* dst, int ld, v8f c) {
  const int l = lane_id(), half = l >> 4, lm = l & 15;
#pragma unroll
  for (int r = 0; r < 8; ++r) dst[(size_t)(r + 8 * half) * ld + lm] = c[r];
}
__device__ __forceinline__ v8f frag_load(const float* src, int ld) {
  const int l = lane_id(), half = l >> 4, lm = l & 15;
  v8f c;
#pragma unroll
  for (int r = 0; r < 8; ++r) c[r] = src[(size_t)(r + 8 * half) * ld + lm];
  return c;
}

// C += A(16x16) * B(16x16) using 4 chained V_WMMA_F32_16X16X4_F32.
template <bool TA, bool TB>
__device__ __forceinline__ v8f mm16(const float* __restrict__ A, int lda,
                                    const float* __restrict__ B, int ldb, v8f c) {
  const int l = lane_id(), half = l >> 4, lm = l & 15;
#pragma unroll
  for (int kk = 0; kk < 4; ++kk) {
    v2f a, b;
#pragma unroll
    for (int r = 0; r < 2; ++r) {
      const int K = kk * 4 + half * 2 + r;
      a[r] = TA ? A[(size_t)K * lda + lm] : A[(size_t)lm * lda + K];
      b[r] = TB ? B[(size_t)lm * ldb + K] : B[(size_t)K * ldb + lm];
    }
    c = __builtin_amdgcn_wmma_f32_16x16x4_f32(false, a, false, b, (short)0, c,
                                              false, false);
  }
  return c;
}

// ---- CDNA5 async copy: 1KB tile (32B/lane) global -> LDS, tracked by ASYNCcnt ----
__device__ __forceinline__ void async_fetch_tile(const float* gbase, unsigned byte_off,
                                                 unsigned lds_off) {
  asm volatile("global_load_async_to_lds_b128 %0, %1, %2 offset:0"
               :: "v"(lds_off), "v"(byte_off), "s"(gbase) : "memory");
  asm volatile("global_load_async_to_lds_b128 %0, %1, %2 offset:16"
               :: "v"(lds_off), "v"(byte_off), "s"(gbase) : "memory");
}
__device__ __forceinline__ void wait_async() {
  asm volatile("s_wait_asynccnt 0x0" ::: "memory");
}

// ---- cooperative 16x16 Gauss-Jordan inverse (augmented in LDS), returns log|det| ----
__device__ float gj_invert(float (*GJ)[LDG]) {
  const int l = lane_id();
  __shared__ float gj_fac[DN];
  if (l < DN) {
#pragma unroll
    for (int j = 0; j < DN; ++j) GJ[l][DN + j] = (l == j) ? 1.0f : 0.0f;
  }
  __syncthreads();
  float ld = 0.0f;
#pragma unroll 1
  for (int k = 0; k < DN; ++k) {
    const float piv = GJ[k][k];
    ld += logf(fabsf(piv) + 1e-30f);
    const float pinv = 1.0f / piv;
    GJ[k][l] *= pinv;  // lane l scales column l of pivot row (32 cols)
    __syncthreads();
    if (l < DN) gj_fac[l] = GJ[l][k];
    __syncthreads();
#pragma unroll 4
    for (int i = 0; i < DN; ++i)
      if (i != k) GJ[i][l] -= gj_fac[i] * GJ[k][l];
    __syncthreads();
  }
  return ld;  // inverse now in columns 16..31
}

// ---- cooperative 16x16 Cholesky in LDS (lower), upper zeroed ----
__device__ void chol16(float (*C)[LDP]) {
  const int l = lane_id();
#pragma unroll 1
  for (int k = 0; k < DN; ++k) {
    __syncthreads();
    const float d = sqrtf(fmaxf(C[k][k], 1e-20f));
    __syncthreads();
    if (l == k) C[k][k] = d;
    if (l > k && l < DN) C[l][k] *= (1.0f / d);
    __syncthreads();
    if (l > k && l < DN) {
      const float cik = C[l][k];
      for (int j = k + 1; j <= l; ++j) C[l][j] -= cik * C[j][k];
    }
    __syncthreads();
  }
  if (l < DN)
    for (int j = l + 1; j < DN; ++j) C[l][j] = 0.0f;
  __syncthreads();
}

__device__ __forceinline__ void matvec16(const float* M, int ld, const float* x,
                                         float* y) {
  const int l = lane_id();
  __syncthreads();
  if (l < DN) {
    float s = 0.0f;
    for (int k = 0; k < DN; ++k) s += M[(size_t)l * ld + k] * x[k];
    y[l] = s;
  }
  __syncthreads();
}

__device__ __forceinline__ float dot16(const float* a, const float* b) {
  float s = 0.0f;
  for (int k = 0; k < DN; ++k) s += a[k] * b[k];
  return s;
}

// deterministic hash -> Box-Muller normal
__device__ __forceinline__ unsigned hashu(unsigned x) {
  x ^= x >> 16; x *= 0x7feb352dU; x ^= x >> 15; x *= 0x846ca68bU; x ^= x >> 16;
  return x;
}
__device__ __forceinline__ float rng_normal(int b, int t, int i) {
  const unsigned s = ((unsigned)b * 1009u + (unsigned)t) * 31u + (unsigned)i;
  const unsigned h1 = hashu(s ^ 0x9E3779B9u);
  const unsigned h2 = hashu(s + 0x85EBCA6Bu);
  const float u1 = ((h1 >> 8) + 1.0f) * (1.0f / 16777218.0f);
  const float u2 = (h2 >> 8) * (1.0f / 16777216.0f);
  return sqrtf(-2.0f * logf(u1)) * cosf(6.28318530718f * u2);
}

// ws constant layout (floats): tau@0 lam@256 AtL@512 AtLA@768 AtLb@1024 Lb@1040
// tau_mu@1056 trans_const@1072 init_const@1073
__global__ void lds_prep_kernel(const float* __restrict__ loc,
                                const float* __restrict__ Tau_p,
                                const float* __restrict__ Lam_p,
                                const float* __restrict__ X,
                                float* __restrict__ cw) {
  __shared__ float sA[DN][LDP], sB[DN][LDP], sC[DN][LDP];
  __shared__ float GJ[DN][LDG];
  __shared__ float svec[DN], svec2[DN], sb[DN], sloc[DN];
  const int l = lane_id();
  const v8f fDiag = diag_mask();

  if (l < DN) {
    for (int j = 0; j < DN; ++j) sA[l][j] = Tau_p[l * DN + j];
    sloc[l] = loc[l];
  }
  __syncthreads();
  v8f ft = mm16<false, true>(&sA[0][0], LDP, &sA[0][0], LDP, vzero8());  // Tau Tau^T
#pragma unroll
  for (int r = 0; r < 8; ++r) ft[r] += fDiag[r] * 1e-8f;
  frag_store(cw + 0, DN, ft);
  frag_store(&sC[0][0], LDP, ft);
  matvec16(&sC[0][0], LDP, sloc, svec);  // tau_mu
  if (l < DN) {
    cw[1056 + l] = svec[l];
    for (int j = 0; j < DN; ++j) GJ[l][j] = sC[l][j];
  }
  __syncthreads();
  const float ld_tau = gj_invert(GJ);
  if (l == 0) cw[1073] = -0.5f * dot16(sloc, svec) + 0.5f * ld_tau;
  __syncthreads();

  if (l < DN)
    for (int j = 0; j < DN; ++j) sA[l][j] = Lam_p[l * DN + j];
  __syncthreads();
  v8f fl = mm16<false, true>(&sA[0][0], LDP, &sA[0][0], LDP, vzero8());  // Lam Lam^T
#pragma unroll
  for (int r = 0; r < 8; ++r) fl[r] += fDiag[r] * 1e-8f;
  frag_store(cw + 256, DN, fl);
  frag_store(&sB[0][0], LDP, fl);
  __syncthreads();

  if (l < DN) {
    for (int j = 0; j < DN; ++j) sA[l][j] = X[l * 17 + j];  // A
    sb[l] = X[l * 17 + DN];                                  // b
  }
  __syncthreads();
  v8f fAtL = mm16<true, false>(&sA[0][0], LDP, &sB[0][0], LDP, vzero8());  // A^T lam
  frag_store(cw + 512, DN, fAtL);
  frag_store(&sC[0][0], LDP, fAtL);
  __syncthreads();
  v8f fAtLA = mm16<false, false>(&sC[0][0], LDP, &sA[0][0], LDP, vzero8());
  frag_store(cw + 768, DN, fAtLA);
  matvec16(&sC[0][0], LDP, sb, svec);   // AtLb
  if (l < DN) cw[1024 + l] = svec[l];
  matvec16(&sB[0][0], LDP, sb, svec2);  // Lb
  if (l < DN) {
    cw[1040 + l] = svec2[l];
    for (int j = 0; j < DN; ++j) GJ[l][j] = sB[l][j];
  }
  __syncthreads();
  const float ld_lam = gj_invert(GJ);
  if (l == 0) cw[1072] = -0.5f * dot16(sb, svec2) + 0.5f * ld_lam;
}

__global__ void lds_forward_kernel(const float* __restrict__ Jrd,
                                   const float* __restrict__ hr,
                                   const float* __restrict__ cw,
                                   float* __restrict__ fJs, float* __restrict__ fhs,
                                   float* __restrict__ fin) {
  __shared__ float GJ[DN][LDG];
  __shared__ float sAtL[DN][LDP], sMiAtL[DN][LDP];
  __shared__ float s_m[DN], s_Mim[DN], s_fh[DN], s_predh[DN], s_jr[DN], s_hr[DN];
  __shared__ float sAtLb[DN], sLb[DN];
  const int b = blockIdx.x, l = lane_id(), half = l >> 4, lm = l & 15;
  const v8f fDiag = diag_mask();

  if (l < DN) {
    for (int j = 0; j < DN; ++j) sAtL[l][j] = cw[512 + l * DN + j];
    sAtLb[l] = cw[1024 + l];
    sLb[l] = cw[1040 + l];
    s_jr[l] = Jrd[((size_t)b * TN) * DN + l];
    s_fh[l] = cw[1056 + l] + hr[((size_t)b * TN) * DN + l];
  }
  const v8f fAtLA = frag_load(cw + 768, DN);
  const v8f fLam = frag_load(cw + 256, DN);
  const float trans_c = cw[1072];
  __syncthreads();

  v8f fJ = frag_load(cw + 0, DN);  // tau
#pragma unroll
  for (int r = 0; r < 8; ++r) fJ[r] += fDiag[r] * (s_jr[lm] + 0.5f);
  float lz = 0.0f;

#pragma unroll 1
  for (int t = 1; t < TN; ++t) {
    // emit scan carry (fJ, fh) at index t-1
    float* pJ = fJs + ((size_t)b * TN + (t - 1)) * 256;
    *(v8f*)(pJ + l * 8) = fJ;
    if (l < DN) {
      fhs[((size_t)b * TN + (t - 1)) * DN + l] = s_fh[l];
      if (t + 8 < TN) {  // stream-ahead prefetch of the observation sequence
        __builtin_prefetch(&Jrd[((size_t)b * TN + t + 8) * DN + l], 0, 3);
        __builtin_prefetch(&hr[((size_t)b * TN + t + 8) * DN + l], 0, 3);
      }
    }

    const v8f fM = fJ + fAtLA;  // M = fJ + AtLA
    frag_store(&GJ[0][0], LDG, fM);
    if (l < DN) s_m[l] = s_fh[l] - sAtLb[l];
    __syncthreads();
    const float ldet = gj_invert(GJ);  // Mi in GJ[.][16..31]
    matvec16(&GJ[0][16], LDG, s_m, s_Mim);
    lz += 0.5f * dot16(s_m, s_Mim) - 0.5f * ldet + trans_c;

    const v8f fMiAtL = mm16<false, false>(&GJ[0][16], LDG, &sAtL[0][0], LDP, vzero8());
    frag_store(&sMiAtL[0][0], LDP, fMiAtL);
    __syncthreads();
    const v8f fS = mm16<true, false>(&sAtL[0][0], LDP, &sMiAtL[0][0], LDP, vzero8());
    v8f fPredJ = fLam - fS;  // predJ = lam - AtL^T Mi AtL

    if (l < DN) {
      float s = sLb[l];
      for (int k = 0; k < DN; ++k) s += sAtL[k][l] * s_Mim[k];  // AtL^T Mim
      s_predh[l] = s;
      s_jr[l] = Jrd[((size_t)b * TN + t) * DN + l];
      s_hr[l] = hr[((size_t)b * TN + t) * DN + l];
    }
    __syncthreads();
    fJ = fPredJ;
#pragma unroll
    for (int r = 0; r < 8; ++r) fJ[r] += fDiag[r] * (s_jr[lm] + 0.5f);
    if (l < DN) s_fh[l] = s_predh[l] + s_hr[l];
    __syncthreads();
  }
  float* pf = fin + (size_t)b * 512;
  frag_store(pf, DN, fJ);
  if (l < DN) pf[256 + l] = s_fh[l];
  if (l == 0) pf[272] = lz;
}

__global__ void lds_backward_kernel(const float* __restrict__ Jrd,
                                    const float* __restrict__ hr,
                                    const float* __restrict__ cw,
                                    const float* __restrict__ fJs,
                                    const float* __restrict__ fhs,
                                    const float* __restrict__ fin,
                                    float* __restrict__ out_kl,
                                    float* __restrict__ EXXT,
                                    float* __restrict__ EX,
                                    float* __restrict__ EXXNT) {
  __shared__ float GJ[DN][LDG];
  __shared__ float sAtL[DN][LDP], sG[DN][LDP], sVn[DN][LDP], sTmp[DN][LDP];
  __shared__ float sStage[2][256];  // async double-buffer for fJs tiles
  __shared__ float s_mn[DN], s_mt[DN], s_fh[DN], s_vec[DN], sAtLb[DN];
  __shared__ float red[32];
  const int b = blockIdx.x, l = lane_id(), half = l >> 4, lm = l & 15;
  const v8f fDiag = diag_mask();
  const unsigned st0 = (unsigned)(size_t)&sStage[0][0];
  const unsigned st1 = (unsigned)(size_t)&sStage[1][0];

  // kick off async fetch of the first backward tile while we set up at t = T-1
  async_fetch_tile(fJs, (unsigned)(((size_t)b * TN + (TN - 2)) * 1024) + l * 32,
                   (((TN - 2) & 1) ? st1 : st0) + (unsigned)(l * 32));

  const float* pf = fin + (size_t)b * 512;
  if (l < DN) {
    for (int j = 0; j < DN; ++j) {
      sAtL[l][j] = cw[512 + l * DN + j];
      GJ[l][j] = pf[l * DN + j];  // fJT
    }
    sAtLb[l] = cw[1024 + l];
    s_fh[l] = pf[256 + l];  // fhT
  }
  const v8f fAtLA = frag_load(cw + 768, DN);
  const float lz = pf[272];
  __syncthreads();
  const float ldT = gj_invert(GJ);
  v8f fV = frag_load(&GJ[0][16], LDG);  // VT
  matvec16(&GJ[0][16], LDG, s_fh, s_mn);  // mT
  const float logZ = lz + cw[1073] + 0.5f * dot16(s_fh, s_mn) - 0.5f * ldT;

  float klq = 0.0f, klh = 0.0f;
  {  // outputs at t = T-1
    const float jr = Jrd[((size_t)b * TN + (TN - 1)) * DN + lm];
    v8f e = fV;
#pragma unroll
    for (int r = 0; r < 8; ++r) {
      e[r] += s_mn[r + 8 * half] * s_mn[lm];
      klq += fDiag[r] * (jr + 0.5f) * e[r];
    }
    frag_store(EXXT + ((size_t)b * TN + (TN - 1)) * 256, DN, e);
    if (l < DN) {
      EX[((size_t)b * TN + (TN - 1)) * DN + l] = s_mn[l];
      klh += hr[((size_t)b * TN + (TN - 1)) * DN + l] * s_mn[l];
    }
  }
  __syncthreads();

#pragma unroll 1
  for (int t = TN - 2; t >= 0; --t) {
    frag_store(&sVn[0][0], LDP, fV);  // V_{t+1}
    wait_async();                     // fJs tile for this t is now in LDS
    __syncthreads();
    const float* st = (t & 1) ? &sStage[1][0] : &sStage[0][0];
    v8f fJ;
#pragma unroll
    for (int r = 0; r < 8; ++r) fJ[r] = st[l * 8 + r];
    if (t > 0)  // overlap next tile's HBM latency with this step's compute
      async_fetch_tile(fJs, (unsigned)(((size_t)b * TN + (t - 1)) * 1024) + l * 32,
                       (((t - 1) & 1) ? st1 : st0) + (unsigned)(l * 32));
    if (l < DN) {
      s_fh[l] = fhs[((size_t)b * TN + t) * DN + l];
      if (t > 8) {
        __builtin_prefetch(&Jrd[((size_t)b * TN + t - 8) * DN + l], 0, 3);
        __builtin_prefetch(&hr[((size_t)b * TN + t - 8) * DN + l], 0, 3);
      }
    }
    const v8f fP = fJ + fAtLA;  // P = fJ + AtLA
    frag_store(&GJ[0][0], LDG, fP);
    __syncthreads();
    gj_invert(GJ);  // Pi
    const v8f fG = mm16<false, false>(&GJ[0][16], LDG, &sAtL[0][0], LDP, vzero8());
    frag_store(&sG[0][0], LDP, fG);
    if (l < DN) s_vec[l] = s_fh[l] - sAtLb[l];
    __syncthreads();
    if (l < DN) {  // m_t = Pi (fh - AtLb) + G m_n
      float s = 0.0f;
      for (int k = 0; k < DN; ++k) s += GJ[l][16 + k] * s_vec[k];
      for (int k = 0; k < DN; ++k) s += sG[l][k] * s_mn[k];
      s_mt[l] = s;
    }
    __syncthreads();
    const v8f fC = mm16<false, false>(&sG[0][0], LDP, &sVn[0][0], LDP, vzero8());  // C = G Vn
    frag_store(&sTmp[0][0], LDP, fC);
    {
      v8f e = fC;
#pragma unroll
      for (int r = 0; r < 8; ++r) e[r] += s_mt[r + 8 * half] * s_mn[lm];
      frag_store(EXXNT + ((size_t)b * (TN - 1) + t) * 256, DN, e);
    }
    __syncthreads();
    const v8f fVt =  // V_t = Pi + C G^T
        mm16<false, true>(&sTmp[0][0], LDP, &sG[0][0], LDP, frag_load(&GJ[0][16], LDG));
    const float jr = Jrd[((size_t)b * TN + t) * DN + lm];
    {
      v8f e = fVt;
#pragma unroll
      for (int r = 0; r < 8; ++r) {
        e[r] += s_mt[r + 8 * half] * s_mt[lm];
        klq += fDiag[r] * (jr + 0.5f) * e[r];
      }
      frag_store(EXXT + ((size_t)b * TN + t) * 256, DN, e);
    }
    if (l < DN) {
      EX[((size_t)b * TN + t) * DN + l] = s_mt[l];
      klh += hr[((size_t)b * TN + t) * DN + l] * s_mt[l];
    }
    __syncthreads();
    fV = fVt;
    if (l < DN) s_mn[l] = s_mt[l];
    __syncthreads();
  }
  red[l] = -0.5f * klq + klh;
  __syncthreads();
  if (l == 0) {
    float s = 0.0f;
    for (int i = 0; i < 32; ++i) s += red[i];
    out_kl[b] = s - logZ;
  }
}

__global__ void lds_sample_kernel(const float* __restrict__ EXXT,
                                  const float* __restrict__ EX,
                                  const float* __restrict__ EXXNT,
                                  float* __restrict__ z) {
  __shared__ float GJ[DN][LDG];
  __shared__ float sC[DN][LDP], sK[DN][LDP], sL[DN][LDP];
  __shared__ float s_m0[DN], s_m1[DN], s_x[DN], s_xn[DN], s_eps[DN];
  const int b = blockIdx.x, l = lane_id(), half = l >> 4, lm = l & 15;
  const v8f fDiag = diag_mask();

  if (l < DN) s_m0[l] = EX[((size_t)b * TN) * DN + l];
  __syncthreads();
  {  // x0 = m0 + chol(V0 + jitter I) eps0 ; V0 = EXXT0 - m0 m0^T
    v8f e = frag_load(EXXT + (size_t)b * TN * 256, DN);
#pragma unroll
    for (int r = 0; r < 8; ++r)
      e[r] += fDiag[r] * JIT - s_m0[r + 8 * half] * s_m0[lm];
    frag_store(&sL[0][0], LDP, e);
  }
  __syncthreads();
  chol16(sL);
  if (l < DN) s_eps[l] = rng_normal(b, 0, l);
  __syncthreads();
  if (l < DN) {
    float s = s_m0[l];
    for (int k = 0; k <= l; ++k) s += sL[l][k] * s_eps[k];
    s_x[l] = s;
    z[((size_t)b * TN) * DN + l] = s;
  }
  __syncthreads();

#pragma unroll 1
  for (int t = 0; t < TN - 1; ++t) {
    // prefetch next iteration's tiles into cache
    if (t + 2 < TN)
      __builtin_prefetch(EXXT + ((size_t)b * TN + t + 2) * 256 + l * 8, 0, 3);
    if (t + 1 < TN - 1)
      __builtin_prefetch(EXXNT + ((size_t)b * (TN - 1) + t + 1) * 256 + l * 8, 0, 3);
    if (l < DN) {
      s_m0[l] = EX[((size_t)b * TN + t) * DN + l];
      s_m1[l] = EX[((size_t)b * TN + t + 1) * DN + l];
    }
    __syncthreads();
    v8f ev = frag_load(EXXT + ((size_t)b * TN + t) * 256, DN);        // -> V_t
    v8f ec = frag_load(EXXNT + ((size_t)b * (TN - 1) + t) * 256, DN); // -> C_t
#pragma unroll
    for (int r = 0; r < 8; ++r) {
      const int M = r + 8 * half;
      ev[r] -= s_m0[M] * s_m0[lm];
      ec[r] -= s_m0[M] * s_m1[lm];
    }
    frag_store(&GJ[0][0], LDG, ev);
    frag_store(&sC[0][0], LDP, ec);
    __syncthreads();
    gj_invert(GJ);  // V_t^{-1}
    const v8f fK = mm16<false, false>(&GJ[0][16], LDG, &sC[0][0], LDP, vzero8());
    frag_store(&sK[0][0], LDP, fK);
    __syncthreads();
    const v8f fS = mm16<true, false>(&sC[0][0], LDP, &sK[0][0], LDP, vzero8());
    v8f en = frag_load(EXXT + ((size_t)b * TN + t + 1) * 256, DN);
#pragma unroll
    for (int r = 0; r < 8; ++r)  // condV = V_{t+1} - C^T K + jitter I
      en[r] += fDiag[r] * JIT - s_m1[r + 8 * half] * s_m1[lm] - fS[r];
    frag_store(&sL[0][0], LDP, en);
    __syncthreads();
    chol16(sL);
    if (l < DN) s_eps[l] = rng_normal(b, t + 1, l);
    __syncthreads();
    if (l < DN) {
      float s = s_m1[l];
      for (int k = 0; k < DN; ++k) s += sK[k][l] * (s_x[k] - s_m0[k]);  // K^T (x - m0)
      for (int k = 0; k <= l; ++k) s += sL[l][k] * s_eps[k];
      s_xn[l] = s;
    }
    __syncthreads();
    if (l < DN) {
      s_x[l] = s_xn[l];
      z[((size_t)b * TN + t + 1) * DN + l] = s_xn[l];
    }
    __syncthreads();
  }
}

extern "C" void kernel_launch(void* const* d_in, const int* in_sizes, int n_in,
                              void* d_out, int out_size, void* d_ws, size_t ws_size,
                              hipStream_t stream) {
  (void)in_sizes; (void)n_in; (void)out_size;
  const float* Jrd = (const float*)d_in[0];
  const float* hr  = (const float*)d_in[1];
  const float* loc = (const float*)d_in[2];
  const float* Tau = (const float*)d_in[3];
  const float* Lam = (const float*)d_in[4];
  const float* X   = (const float*)d_in[5];
  float* out = (float*)d_out;
  float* cw  = (float*)d_ws;

  const size_t FIN_OFF = 2048;
  const size_t FJS_OFF = FIN_OFF + (size_t)BN * 512;
  const size_t FHS_OFF = FJS_OFF + (size_t)BN * TN * 256;
  const size_t WS_FLOATS = FHS_OFF + (size_t)BN * TN * DN;
  if (ws_size < WS_FLOATS * sizeof(float)) return;  // scratch too small

  float* fin = cw + FIN_OFF;
  float* fJs = cw + FJS_OFF;
  float* fhs = cw + FHS_OFF;

  // d_out = concat(z, kl, EXXT, EX, EXXNT)
  float* z     = out;
  float* kl    = out + (size_t)BN * TN * DN;
  float* EXXT  = kl + BN;
  float* EX    = EXXT + (size_t)BN * TN * 256;
  float* EXXNT = EX + (size_t)BN * TN * DN;

  lds_prep_kernel<<<1, 32, 0, stream>>>(loc, Tau, Lam, X, cw);
  lds_forward_kernel<<<BN, 32, 0, stream>>>(Jrd, hr, cw, fJs, fhs, fin);
  lds_backward_kernel<<<BN, 32, 0, stream>>>(Jrd, hr, cw, fJs, fhs, fin, kl,
                                             EXXT, EX, EXXNT);
  lds_sample_kernel<<<BN, 32, 0, stream>>>(EXXT, EX, EXXNT, z);
}